// SLayerRational_43190191128606
// MI455X (gfx1250) — compile-verified
//
#include <hip/hip_runtime.h>
#include <math.h>

typedef __attribute__((ext_vector_type(2))) float v2f;
typedef __attribute__((ext_vector_type(8))) float v8f;

namespace {
constexpr int kB = 128;
constexpr int kP = 2048;
constexpr int kN = 256;
constexpr int kThreads = 512; // 16 waves * 32 lanes; 16 center-tiles of 16 -> N=256
}

// Fast hardware transcendentals: v_rcp_f32, v_log_f32 (log2), v_exp_f32 (exp2).
// Fallbacks use libm names that HIP declares with __device__ overloads, so the
// host semantic-check pass also resolves them.
__device__ __forceinline__ float fast_rcp(float x) {
#if __has_builtin(__builtin_amdgcn_rcpf)
  return __builtin_amdgcn_rcpf(x);
#else
  return 1.0f / x;
#endif
}
__device__ __forceinline__ float fast_log2(float x) {
#if __has_builtin(__builtin_amdgcn_logf)
  return __builtin_amdgcn_logf(x);
#else
  return log2f(x);
#endif
}
__device__ __forceinline__ float fast_exp2(float x) {
#if __has_builtin(__builtin_amdgcn_exp2f)
  return __builtin_amdgcn_exp2f(x);
#else
  return exp2f(x);
#endif
}

// Core loop: wave-level 16-center tile, WMMA f32 16x16x4 with all-ones B performs
// the sum over points in fp32 matrix hardware. Two accumulators break the
// D->C RAW chain between consecutive WMMAs.
template <bool POWGEN, int PCHUNK>
__device__ __forceinline__ v8f
accumulate_chunk(const float4* __restrict__ pts, float c0, float c1,
                 float w0, float w1, float e, int khalf) {
  v8f acc0 = {};
  v8f acc1 = {};
  const v2f ones = {1.0f, 1.0f};
  for (int pg = 0; pg < PCHUNK; pg += 8) {
#pragma unroll
    for (int g = 0; g < 2; ++g) {
      const int p = pg + g * 4 + khalf; // lanes 0-15: K=0,1  lanes 16-31: K=2,3
      const float4 q0 = pts[p];
      const float4 q1 = pts[p + 1];
      float s0 = fabsf(c0 - q0.x) * w0 + fabsf(c1 - q0.y) * w1;
      float s1 = fabsf(c0 - q1.x) * w0 + fabsf(c1 - q1.y) * w1;
      if (POWGEN) { // s >= 0: s^e = exp2(e*log2(s)); s==0 -> -inf -> 0 for e>0
        s0 = fast_exp2(e * fast_log2(s0));
        s1 = fast_exp2(e * fast_log2(s1));
      }
      v2f a;
      a.x = q0.z * fast_rcp(1.0f + s0); // mask folded into the numerator
      a.y = q1.z * fast_rcp(1.0f + s1);
      if (g == 0)
        acc0 = __builtin_amdgcn_wmma_f32_16x16x4_f32(
            false, a, false, ones, (short)0, acc0, false, false);
      else
        acc1 = __builtin_amdgcn_wmma_f32_16x16x4_f32(
            false, a, false, ones, (short)0, acc1, false, false);
    }
  }
  return acc0 + acc1;
}

template <int SPLIT>
__global__ __launch_bounds__(kThreads)
void slayer_partial(const float* __restrict__ batch,
                    const float* __restrict__ not_dummy,
                    const float* __restrict__ centers,
                    const float* __restrict__ sharp,
                    const float* __restrict__ expo,
                    float* __restrict__ part) { // [SPLIT][kB][kN]
  constexpr int PCHUNK = kP / SPLIT;
  __shared__ float4 pts[PCHUNK]; // (x0, x1, mask, pad)

  const int b = blockIdx.x;
  const int sidx = blockIdx.y;
  const int tid = (int)threadIdx.x;
  const int lane = tid & 31;
  const int wave = tid >> 5;
  const int pbase = sidx * PCHUNK;

  // Cooperative staging of this chunk's points + mask into LDS.
  for (int i = tid; i < PCHUNK; i += kThreads) {
    const int p = pbase + i;
    const float2 x =
        *reinterpret_cast<const float2*>(batch + ((size_t)b * kP + p) * 2);
    const float msk = not_dummy[(size_t)b * kP + p];
    pts[i] = make_float4(x.x, x.y, msk, 0.0f);
  }

  // Per-lane center (wave owns a 16-center tile; A-matrix rows M = lane&15).
  const int n = wave * 16 + (lane & 15);
  const float c0 = centers[n * 2 + 0];
  const float c1 = centers[n * 2 + 1];
  const float w0 = fabsf(sharp[n * 2 + 0]);
  const float w1 = fabsf(sharp[n * 2 + 1]);
  const float e = expo[0];
  const int khalf = (lane >> 4) << 1; // 0 for lanes 0-15, 2 for lanes 16-31

  __syncthreads();

  // Uniform branch (e identical across lanes) -> EXEC is all-1s inside,
  // as WMMA requires.
  v8f acc;
  if (e == 1.0f)
    acc = accumulate_chunk<false, PCHUNK>(pts, c0, c1, w0, w1, e, khalf);
  else
    acc = accumulate_chunk<true, PCHUNK>(pts, c0, c1, w0, w1, e, khalf);

  // D tile layout: VGPR j -> M=j (lanes 0-15) / M=8+j (lanes 16-31);
  // all 16 columns are identical (B was all ones), so lanes 0 and 16 emit
  // the 16 per-center sums of this tile.
  float* dst = part + ((size_t)sidx * kB + b) * kN + wave * 16;
  if (lane == 0) {
#pragma unroll
    for (int j = 0; j < 8; ++j) dst[j] = acc[j];
  } else if (lane == 16) {
#pragma unroll
    for (int j = 0; j < 8; ++j) dst[8 + j] = acc[j];
  }
}

template <int SPLIT>
__global__ void slayer_reduce(const float* __restrict__ part,
                              float* __restrict__ out) {
  const int i = blockIdx.x * blockDim.x + threadIdx.x;
  if (i < kB * kN) {
    float a = 0.0f;
#pragma unroll
    for (int s = 0; s < SPLIT; ++s)
      a += part[(size_t)s * (kB * kN) + i];
    out[i] = a;
  }
}

extern "C" void kernel_launch(void* const* d_in, const int* in_sizes, int n_in,
                              void* d_out, int out_size, void* d_ws,
                              size_t ws_size, hipStream_t stream) {
  const float* batch     = (const float*)d_in[0]; // (B, P, 2)
  const float* not_dummy = (const float*)d_in[1]; // (B, P)
  const float* centers   = (const float*)d_in[2]; // (N, 2)
  const float* sharp     = (const float*)d_in[3]; // (N, 2)
  const float* expo      = (const float*)d_in[4]; // (1,)
  float* out = (float*)d_out;                     // (B, N)

  constexpr int kSplit = 8;
  const size_t need = (size_t)kSplit * kB * kN * sizeof(float);

  if (ws_size >= need) {
    float* part = (float*)d_ws;
    slayer_partial<kSplit><<<dim3(kB, kSplit), kThreads, 0, stream>>>(
        batch, not_dummy, centers, sharp, expo, part);
    slayer_reduce<kSplit><<<(kB * kN + 255) / 256, 256, 0, stream>>>(part, out);
  } else {
    // Fallback: single split writes the final sums directly (no workspace).
    slayer_partial<1><<<dim3(kB, 1), kThreads, 0, stream>>>(
        batch, not_dummy, centers, sharp, expo, out);
  }
}